// MixtureAlignmentLogLikelihood_83305185673806
// MI455X (gfx1250) — compile-verified
//
#include <hip/hip_runtime.h>
#include <hip/hip_bf16.h>
#include <math.h>

// ---------------------------------------------------------------------------
// Problem shape (fixed by setup_inputs): B=256, T=2048, L=64, K=64, fp32.
// dot[b,t] = (S_b/64) * rowsum_softmax(b,t);  out = sum scale*log(dot).
// Kernel 1: stream trg_sent once (async global->LDS, double buffered),
//           softmax rowsums via v_wmma_f32_16x16x4_f32 (A = p-tile, B = ones),
//           write rowsum[B,T] + atomic per-batch partial S.
// Kernel 2: segment scales via LDS scans + log reduction.
// ---------------------------------------------------------------------------

typedef __attribute__((ext_vector_type(2))) float v2f;
typedef __attribute__((ext_vector_type(8))) float v8f;
typedef __attribute__((ext_vector_type(4))) int   v4i;

#define AS1 __attribute__((address_space(1)))
#define AS3 __attribute__((address_space(3)))

constexpr int Bb = 256;
constexpr int Tt = 2048;
constexpr int Kk = 64;

constexpr int TILE_ROWS  = 64;               // rows per tile (4 waves * 16 rows)
constexpr int ROW_PAD_DW = 68;               // 64 data dwords + 4 pad -> conflict-free
constexpr int ROW_BYTES  = ROW_PAD_DW * 4;   // 272
constexpr int NTHREADS1  = 128;              // 4 waves (wave32)
constexpr int TILES_PER_BLOCK = 8;           // 512 rows per block
constexpr int ROWS_PER_BLOCK  = TILE_ROWS * TILES_PER_BLOCK;

#if __has_builtin(__builtin_amdgcn_global_load_async_to_lds_b128) && \
    __has_builtin(__builtin_amdgcn_s_wait_asynccnt)
#define USE_ASYNC_LDS 1
#else
#define USE_ASYNC_LDS 0
#endif

__device__ __forceinline__ void issue_tile(const float* __restrict__ g,
                                           float* ldsbuf, int tid) {
  // Copy TILE_ROWS x 64 floats (16 KB) into padded LDS rows; 8 x b128 per lane.
#pragma unroll
  for (int r = 0; r < 8; ++r) {
    int o   = r * NTHREADS1 + tid;        // 16-byte chunk index, 0..1023
    int row = o >> 4;                     // 16 chunks per row
    int ch  = o & 15;
#if USE_ASYNC_LDS
    AS1 v4i* gp = (AS1 v4i*)(const_cast<float*>(g) + (size_t)o * 4);
    AS3 v4i* lp = (AS3 v4i*)((char*)ldsbuf + row * ROW_BYTES + ch * 16);
    __builtin_amdgcn_global_load_async_to_lds_b128(gp, lp, 0, 0);
#else
    float4 v = *(const float4*)(g + (size_t)o * 4);
    *(float4*)((char*)ldsbuf + row * ROW_BYTES + ch * 16) = v;
#endif
  }
}

__device__ __forceinline__ void wait_async_le(int pending) {
#if USE_ASYNC_LDS
  if (pending > 0) __builtin_amdgcn_s_wait_asynccnt(8);
  else             __builtin_amdgcn_s_wait_asynccnt(0);
#else
  (void)pending;
#endif
}

// Process 16 rows owned by this wave; returns sum of the 16 rowsums
// (identical on every lane) and stores the 16 rowsums to rowsum_out.
__device__ __forceinline__ float compute_rows(const float* ldsbuf, int lane,
                                              int wave,
                                              float* __restrict__ rowsum_out) {
  const int row = (wave << 4) + (lane & 15);
  const int h   = lane >> 4;  // 0: k%4 in {0,1}; 1: k%4 in {2,3}  (WMMA A layout)
  const float* rowbase = ldsbuf + row * ROW_PAD_DW;

  float2 v[16];
#pragma unroll
  for (int c = 0; c < 16; ++c)
    v[c] = *(const float2*)(rowbase + 4 * c + 2 * h);

  // row max (this lane holds 32 of the 64 values; partner lane holds the rest)
  float mx = v[0].x;
#pragma unroll
  for (int c = 0; c < 16; ++c) mx = fmaxf(mx, fmaxf(v[c].x, v[c].y));
  mx = fmaxf(mx, __shfl_xor(mx, 16, 32));

  // exp and sum
  float s = 0.f;
#pragma unroll
  for (int c = 0; c < 16; ++c) {
    float e0 = __expf(v[c].x - mx);
    float e1 = __expf(v[c].y - mx);
    v[c].x = e0; v[c].y = e1;
    s += e0 + e1;
  }
  s += __shfl_xor(s, 16, 32);
  float inv = 1.0f / s;

  // rowsum of softmax via matrix unit: C += A(16x4 p-chunk) x ones(4x16).
  v8f C = {};
  v2f ones; ones.x = 1.0f; ones.y = 1.0f;
#pragma unroll
  for (int c = 0; c < 16; ++c) {
    v2f a; a.x = v[c].x * inv; a.y = v[c].y * inv;
    C = __builtin_amdgcn_wmma_f32_16x16x4_f32(false, a, false, ones,
                                              (short)0, C, false, false);
  }
  // C[r], lanes 0-15 -> rowsum[M=r]; lanes 16-31 -> rowsum[M=8+r].
  if ((lane & 15) == 0) {
    float4 p0 = {C[0], C[1], C[2], C[3]};
    float4 p1 = {C[4], C[5], C[6], C[7]};
    float* dst = rowsum_out + (wave << 4) + h * 8;
    *(float4*)dst       = p0;
    *(float4*)(dst + 4) = p1;
  }
  float part = C[0] + C[1] + C[2] + C[3] + C[4] + C[5] + C[6] + C[7];
  part += __shfl_xor(part, 16, 32);  // all 16 rowsums of this wave's rows
  return part;
}

__global__ void __launch_bounds__(NTHREADS1)
k1_softmax_rowsum(const float* __restrict__ trg, float* __restrict__ rowsum,
                  float* __restrict__ Spart) {
  __shared__ alignas(16) float lds[2][TILE_ROWS * ROW_PAD_DW];

  const int tid  = threadIdx.x;
  const int lane = tid & 31;
  const int wave = tid >> 5;

  const long long row0 = (long long)blockIdx.x * ROWS_PER_BLOCK;
  const float* gbase   = trg + row0 * Kk;
  const int b          = (int)(row0 >> 11);  // / T

  issue_tile(gbase, lds[0], tid);

  float acc = 0.f;
  for (int t = 0; t < TILES_PER_BLOCK; ++t) {
    if (t + 1 < TILES_PER_BLOCK)
      issue_tile(gbase + (size_t)(t + 1) * TILE_ROWS * Kk, lds[(t + 1) & 1], tid);
    wait_async_le(t + 1 < TILES_PER_BLOCK ? 8 : 0);
    __syncthreads();  // tile t resident for all waves
    acc += compute_rows(lds[t & 1], lane, wave,
                        rowsum + row0 + (size_t)t * TILE_ROWS);
    __syncthreads();  // buffer t&1 free for tile t+2
  }
  if (lane == 0) atomicAdd(&Spart[b], acc);
}

// ---------------------------------------------------------------------------
__global__ void k0_init(float* Spart, float* out) {
  int i = threadIdx.x;
  if (i < Bb) Spart[i] = 0.f;
  if (i == 0) out[0] = 0.f;
}

__global__ void __launch_bounds__(256)
k2_scales_reduce(const int* __restrict__ bnd, const float* __restrict__ rowsum,
                 const float* __restrict__ Spart, float* __restrict__ out) {
  __shared__ int sa[Tt];  // cummax of (eff ? t : 0)
  __shared__ int sb[Tt];  // reverse cummin of (b ? t : INF)
  const int b = blockIdx.x, tid = threadIdx.x;
  const int INFV = 1 << 20;

  for (int t = tid; t < Tt; t += 256) {
    int f = bnd[b * Tt + t] > 0;
    sa[t] = (f || t == 0) ? t : 0;
    sb[t] = f ? t : INFV;
  }
  __syncthreads();

  for (int off = 1; off < Tt; off <<= 1) {
    int va[8], vb[8];
#pragma unroll
    for (int j = 0; j < 8; ++j) {
      int t = tid + j * 256;
      va[j] = (t >= off) ? sa[t - off] : 0;
      vb[j] = (t + off < Tt) ? sb[t + off] : INFV;
    }
    __syncthreads();
#pragma unroll
    for (int j = 0; j < 8; ++j) {
      int t = tid + j * 256;
      sa[t] = max(sa[t], va[j]);
      sb[t] = min(sb[t], vb[j]);
    }
    __syncthreads();
  }

  const float phi = Spart[b] * (1.0f / (float)Tt) * (1.0f / (float)Kk);
  float acc = 0.f;
  for (int t = tid; t < Tt; t += 256) {
    int st = sa[t];
    int en = (t + 1 < Tt) ? sb[t + 1] : INFV;
    float scale = (en < INFV) ? 1.0f / (float)max(en - st, 1) : 1.0f;
    float dot = phi * rowsum[(size_t)b * Tt + t];
    acc += scale * __logf(dot);  // dot > 0 always (softmax); EPS==0 in fp32
  }
#pragma unroll
  for (int o = 16; o >= 1; o >>= 1) acc += __shfl_xor(acc, o, 32);
  __shared__ float wsum[8];
  if ((tid & 31) == 0) wsum[tid >> 5] = acc;
  __syncthreads();
  if (tid == 0) {
    float s = 0.f;
#pragma unroll
    for (int i = 0; i < 8; ++i) s += wsum[i];
    atomicAdd(out, s);
  }
}

// ---------------------------------------------------------------------------
extern "C" void kernel_launch(void* const* d_in, const int* in_sizes, int n_in,
                              void* d_out, int out_size, void* d_ws,
                              size_t ws_size, hipStream_t stream) {
  // inputs: 0 src_sent (unused), 1 trg_sent, 2 src_boundary (unused), 3 trg_boundary
  const float* trg = (const float*)d_in[1];
  const int* tbnd  = (const int*)d_in[3];
  float* out       = (float*)d_out;

  float* Spart  = (float*)d_ws;           // B floats
  float* rowsum = (float*)d_ws + 1024;    // B*T floats (2 MB), 4 KB offset

  k0_init<<<1, 256, 0, stream>>>(Spart, out);

  const int nblk1 = (Bb * Tt) / ROWS_PER_BLOCK;  // 1024
  k1_softmax_rowsum<<<nblk1, NTHREADS1, 0, stream>>>(trg, rowsum, Spart);

  k2_scales_reduce<<<Bb, 256, 0, stream>>>(tbnd, rowsum, Spart, out);
}